// CONV_A_64115271795341
// MI455X (gfx1250) — compile-verified
//
#include <hip/hip_runtime.h>

typedef __attribute__((ext_vector_type(2))) float v2f;
typedef __attribute__((ext_vector_type(4))) float v4f;
typedef __attribute__((ext_vector_type(8))) float v8f;
typedef __attribute__((ext_vector_type(4))) int   v4i;

#define GAS __attribute__((address_space(1)))
#define LAS __attribute__((address_space(3)))

#define BN 8
#define CI 64
#define HH 128
#define WW 128
#define DOUT 64
#define LDSW 132        // padded LDS row stride (floats)
#define PHC 32          // channels staged per phase
#define SXR 4           // staged rows per channel (2 output rows + halo)
#define NCHUNK 144      // K=576 split into chunks of 4
#define SX_FLOATS (PHC * SXR * LDSW)    // 16896
#define SA_FLOATS (4 * 72 * 32 * 2)     // 18432 (A frags for one phase)
#define SMEM_BYTES ((SX_FLOATS + SA_FLOATS) * 4)   // 141312 B dynamic LDS

// -------- CDNA5 async-to-LDS path (builtins confirmed present) --------
#if defined(__has_builtin)
#if __has_builtin(__builtin_amdgcn_global_load_async_to_lds_b128) && \
    __has_builtin(__builtin_amdgcn_global_load_async_to_lds_b32) &&  \
    __has_builtin(__builtin_amdgcn_s_wait_asynccnt)
#define USE_ASYNC 1
#endif
#endif
#ifndef USE_ASYNC
#define USE_ASYNC 0
#endif

// ---------------------------------------------------------------------------
// Prep: fused effective weight in WMMA A-fragment order.
//   W_eff[c,t,d] = w[c,t,d] - (1/9)*sum_j w[c,j,d] + (1/9)*w_lin[d,c]
// afrag index: (((mt*NCHUNK + qlin)*32 + lane)*2 + j), K-in-chunk = 2*hi + j
// ---------------------------------------------------------------------------
__global__ __launch_bounds__(256) void ca_build_afrag(const float* __restrict__ wconv,
                                                      const float* __restrict__ wlin,
                                                      float* __restrict__ afrag) {
    int idx = blockIdx.x * 256 + threadIdx.x;
    if (idx >= 4 * NCHUNK * 32) return;
    int lane = idx & 31;
    int qlin = (idx >> 5) % NCHUNK;
    int mt   = idx / (NCHUNK * 32);
    int p    = qlin / 72;
    int rm   = qlin % 72;
    int t    = rm >> 3;
    int cq   = rm & 7;
    int d    = mt * 16 + (lane & 15);
    int hi   = lane >> 4;
    #pragma unroll
    for (int j = 0; j < 2; ++j) {
        int c = p * PHC + cq * 4 + hi * 2 + j;
        float s = 0.f;
        #pragma unroll
        for (int tt = 0; tt < 9; ++tt) s += wconv[(c * 9 + tt) * DOUT + d];
        float v = wconv[(c * 9 + t) * DOUT + d] - s * (1.f / 9.f)
                + wlin[d * CI + c] * (1.f / 9.f);
        afrag[(size_t)idx * 2 + j] = v;
    }
}

// ---------------------------------------------------------------------------
// Main: implicit-GEMM 3x3 conv via V_WMMA_F32_16X16X4_F32.
// WG per (b, row-pair): M=64, N=256 (2 rows x 128), K=576.
// 8 waves: wave -> (row rsel = wv>>2, col base = (wv&3)*32), 2 N-tiles,
// 4 M-tiles each => 8 WMMAs per A-fragment load (A bytes/FLOP halved).
// Per phase (32 ch): stage 4 x rows + phase A frags in LDS via
// GLOBAL_LOAD_ASYNC_TO_LDS (ASYNCcnt), then s_wait_asynccnt + barrier.
// ---------------------------------------------------------------------------
__global__ __launch_bounds__(256) void ca_conv_wmma(const float* __restrict__ x,
                                                    const float* __restrict__ afrag,
                                                    float* __restrict__ out) {
    extern __shared__ float smem[];
    float* sx = smem;                 // [32ch][4 rows][132]
    float* sa = smem + SX_FLOATS;     // [4 mt][72 chunks][32 lanes][2]

    const int wg    = blockIdx.x;     // 0..511
    const int b     = wg >> 6;
    const int h0    = (wg & 63) * 2;  // first of two output rows
    const int tid   = threadIdx.x;
    const int lane  = tid & 31;
    const int wv    = tid >> 5;
    const int rsel  = wv >> 2;        // which output row (0/1)
    const int colb  = (wv & 3) * 32;  // column base (two 16-px tiles)
    const int laneN = lane & 15;
    const int hi    = lane >> 4;

    v8f acc[4][2];
    #pragma unroll
    for (int mt = 0; mt < 4; ++mt)
        #pragma unroll
        for (int tl = 0; tl < 2; ++tl) acc[mt][tl] = (v8f){};

    for (int p = 0; p < 2; ++p) {
        // ---- stage A fragments for this phase: 4 blocks of 288 float4 ----
        #pragma unroll
        for (int mt = 0; mt < 4; ++mt) {
            const v4f* src = (const v4f*)(afrag + (size_t)(mt * NCHUNK + p * 72) * 64);
            v4f*       dst = (v4f*)(sa + (size_t)mt * 4608);
            #pragma unroll
            for (int k = 0; k < 2; ++k) {
                int i = tid + k * 256;
                if (i < 288) {
#if USE_ASYNC
                    __builtin_amdgcn_global_load_async_to_lds_b128(
                        (GAS v4i*)(src + i), (LAS v4i*)(dst + i), 0, 0);
#else
                    dst[i] = src[i];
#endif
                }
            }
        }

        // ---- stage x: wave -> channels {wv, wv+8, wv+16, wv+24}, 4 rows, lanes->cols
        #pragma unroll
        for (int ci = 0; ci < 4; ++ci) {
            int cl = wv + ci * 8;
            #pragma unroll
            for (int rr = 0; rr < 4; ++rr) {
                int gh = h0 - 1 + rr;
                gh = gh < 0 ? 0 : (gh > 127 ? 127 : gh);   // edge padding
                const float* xr = x + ((size_t)(b * CI + p * PHC + cl) * HH + gh) * WW;
                float*       sr = sx + (cl * SXR + rr) * LDSW;
                #pragma unroll
                for (int k = 0; k < 5; ++k) {
                    int col = lane + k * 32;               // 0..159
                    if (col < 132) {
                        int gw = col - 1;
                        gw = gw < 0 ? 0 : (gw > 127 ? 127 : gw);
#if USE_ASYNC
                        __builtin_amdgcn_global_load_async_to_lds_b32(
                            (GAS int*)(xr + gw), (LAS int*)(sr + col), 0, 0);
#else
                        sr[col] = xr[gw];
#endif
                    }
                }
            }
        }

#if USE_ASYNC
        __builtin_amdgcn_s_wait_asynccnt(0);
#endif
        __syncthreads();

        // ---- K loop: 72 chunks of 4 for this phase ----
        for (int t = 0; t < 9; ++t) {
            const int r  = t / 3;
            const int dw = (t % 3) - 1;
            #pragma unroll
            for (int cq = 0; cq < 8; ++cq) {
                const int qp = t * 8 + cq;
                // B fragments (4x16 f32): VGPR v, half hi -> K = v + 2*hi
                const int cA   = cq * 4 + hi * 2;
                const int base = (cA * SXR + rsel + r) * LDSW + colb + laneN + 1 + dw;
                v2f B0, B1;
                B0.x = sx[base];
                B0.y = sx[base + SXR * LDSW];              // channel +1 => K+1
                B1.x = sx[base + 16];
                B1.y = sx[base + 16 + SXR * LDSW];

                // A fragments from LDS (lane-contiguous ds_load_b64, no conflicts)
                const float* sap = sa + ((size_t)qp * 32 + lane) * 2;
                #pragma unroll
                for (int mt = 0; mt < 4; ++mt) {
                    v2f A = *(const v2f*)(sap + mt * 4608);
                    acc[mt][0] = __builtin_amdgcn_wmma_f32_16x16x4_f32(
                        false, A, false, B0, (short)0, acc[mt][0], false, false);
                    acc[mt][1] = __builtin_amdgcn_wmma_f32_16x16x4_f32(
                        false, A, false, B1, (short)0, acc[mt][1], false, false);
                }
            }
        }
        __syncthreads();
    }

    // ---- store: C/D layout (16x16 f32): VGPR j -> M = j + 8*hi, N = laneN ----
    const int hrow = h0 + rsel;
    #pragma unroll
    for (int tl = 0; tl < 2; ++tl) {
        float* ob = out + (size_t)b * DOUT * HH * WW + (size_t)hrow * WW
                  + colb + tl * 16 + laneN;
        #pragma unroll
        for (int mt = 0; mt < 4; ++mt) {
            #pragma unroll
            for (int j = 0; j < 8; ++j) {
                int d = mt * 16 + j + 8 * hi;
                ob[(size_t)d * HH * WW] = acc[mt][tl][j];
            }
        }
    }
}

extern "C" void kernel_launch(void* const* d_in, const int* in_sizes, int n_in,
                              void* d_out, int out_size, void* d_ws, size_t ws_size,
                              hipStream_t stream) {
    const float* x  = (const float*)d_in[0];   // (8,64,128,128)
    const float* wc = (const float*)d_in[1];   // (64,9,64)
    const float* wl = (const float*)d_in[2];   // (64,64)
    float* outp  = (float*)d_out;              // (8,64,128,128)
    float* afrag = (float*)d_ws;               // 4*144*32*2*4 = 147456 B

    ca_build_afrag<<<(4 * NCHUNK * 32 + 255) / 256, 256, 0, stream>>>(wc, wl, afrag);
    ca_conv_wmma<<<BN * HH / 2, 256, SMEM_BYTES, stream>>>(x, afrag, outp);
}